// MainModel_8383776162105
// MI455X (gfx1250) — compile-verified
//
#include <hip/hip_runtime.h>
#include <hip/hip_bf16.h>

#define BB   8
#define NS   128
#define NT   128
#define CDIM 64
#define DEG  10
#define NSTEPS 5
#define TAU  0.1f

typedef __attribute__((ext_vector_type(2))) float v2f;
typedef __attribute__((ext_vector_type(8))) float v8f;

// ---------------------------------------------------------------------------
// Kernel 1: neighbor lists. Edge slots for node v are exactly [v*DEG,(v+1)*DEG)
// (src = repeat(arange)); to_dense_adj collapses duplicates; top_k over a 0/1
// row returns distinct neighbors in ascending local-index order. Reproduce via
// selection scan: ascending distinct values of dst%N, plus the count.
// ---------------------------------------------------------------------------
__global__ void nbr_kernel(const int* __restrict__ dst,
                           int* __restrict__ nbr, int* __restrict__ cnt,
                           int nNodes)
{
    int v = blockIdx.x * blockDim.x + threadIdx.x;
    if (v >= nNodes) return;
    int d[DEG];
#pragma unroll
    for (int e = 0; e < DEG; ++e) d[e] = dst[v * DEG + e] & (NS - 1);
    int prev = -1, c = 0;
#pragma unroll
    for (int s = 0; s < DEG; ++s) {
        int cur = 1 << 30;
#pragma unroll
        for (int e = 0; e < DEG; ++e) {
            int x = d[e];
            if (x > prev && x < cur) cur = x;
        }
        bool ok = cur < (1 << 30);
        nbr[v * DEG + s] = ok ? cur : 0;
        if (ok) { prev = cur; ++c; }
    }
    cnt[v] = c;
}

// ---------------------------------------------------------------------------
// Kernel 2: sim[b] = h_s[b] (128x64) @ h_t[b]^T (64x128), fp32 WMMA 16x16x4.
// One wave per 16x16 output tile; 16 WMMAs over K=64.
// A 16x4 f32 frag: lane L -> m=L%16, regs = A[m][2*(L/16)+{0,1}]
// B  4x16 f32 frag: lane L -> n=L%16, regs = B[2*(L/16)+{0,1}][n] = h_t[n][k..]
// C/D 16x16 f32: lane L -> n=L%16, VGPR v -> m = v + 8*(L/16)
// ---------------------------------------------------------------------------
__global__ __launch_bounds__(256)
void sim_kernel(const float* __restrict__ xs, const float* __restrict__ xt,
                float* __restrict__ sim)
{
    int lane = threadIdx.x & 31;
    int wave = threadIdx.x >> 5;
    int b    = blockIdx.x >> 3;                 // 8 blocks per batch
    int tile = ((blockIdx.x & 7) << 3) | wave;  // 0..63
    int mt   = (tile >> 3) << 4;
    int nt   = (tile & 7) << 4;
    int half = lane >> 4;
    int r16  = lane & 15;

    const float* arow = xs + (size_t)(b * NS + mt + r16) * CDIM;
    const float* brow = xt + (size_t)(b * NT + nt + r16) * CDIM;

    v8f c = {};
#pragma unroll
    for (int k0 = 0; k0 < CDIM; k0 += 4) {
        float2 af = *reinterpret_cast<const float2*>(arow + k0 + 2 * half);
        float2 bf = *reinterpret_cast<const float2*>(brow + k0 + 2 * half);
        v2f a; a[0] = af.x; a[1] = af.y;
        v2f bm; bm[0] = bf.x; bm[1] = bf.y;
        c = __builtin_amdgcn_wmma_f32_16x16x4_f32(false, a, false, bm,
                                                  (short)0, c, false, false);
    }

    float* obase = sim + (size_t)b * NS * NT;
#pragma unroll
    for (int v = 0; v < 8; ++v) {
        int m = mt + v + 8 * half;
        int n = nt + r16;
        obase[m * NT + n] = c[v];
    }
}

// ---------------------------------------------------------------------------
// Kernel 3: one block per (b,i), one thread per j. Register-resident 10x10
// masked-Sinkhorn per pair, then block softmax over j.
// ---------------------------------------------------------------------------
__global__ __launch_bounds__(NT)
void pair_kernel(const float* __restrict__ sim,
                 const int* __restrict__ nbrS, const int* __restrict__ cntS,
                 const int* __restrict__ nbrT, const int* __restrict__ cntT,
                 float* __restrict__ out)
{
    __shared__ float srows[DEG][NT];
    __shared__ float red[NT];

    int j = threadIdx.x;
    int b = blockIdx.x >> 7;
    int i = blockIdx.x & (NS - 1);

    int ds = cntS[b * NS + i];
    int isv[DEG];
#pragma unroll
    for (int p = 0; p < DEG; ++p) isv[p] = nbrS[(b * NS + i) * DEG + p];

#pragma unroll
    for (int p = 0; p < DEG; ++p)
        srows[p][j] = (p < ds) ? sim[(size_t)(b * NS + isv[p]) * NT + j] : 0.0f;
    __syncthreads();

    int dt = cntT[b * NT + j];
    int it[DEG];
#pragma unroll
    for (int q = 0; q < DEG; ++q) it[q] = nbrT[(b * NT + j) * DEG + q];

    // ---- masked row softmax over q, divided by tau ----
    float o[DEG][DEG];
#pragma unroll
    for (int p = 0; p < DEG; ++p) {
        bool pv = p < ds;
        float a[DEG];
#pragma unroll
        for (int q = 0; q < DEG; ++q) {
            float sv = srows[p][it[q]];
            a[q] = (pv && (q < dt)) ? sv : -3.0e38f;
        }
        float m = a[0];
#pragma unroll
        for (int q = 1; q < DEG; ++q) m = fmaxf(m, a[q]);
        float sum = 0.0f;
#pragma unroll
        for (int q = 0; q < DEG; ++q) {
            float e = __expf(a[q] - m);
            e = (pv && (q < dt)) ? e : 0.0f;
            o[p][q] = e;
            sum += e;
        }
        float inv = pv ? (1.0f / (sum * TAU)) : 0.0f;
#pragma unroll
        for (int q = 0; q < DEG; ++q) o[p][q] *= inv;
    }

    // ---- Sinkhorn: alternating column / row normalization ----
    for (int s = 0; s < NSTEPS; ++s) {
#pragma unroll
        for (int q = 0; q < DEG; ++q) {
            float cs = 1e-8f;
#pragma unroll
            for (int p = 0; p < DEG; ++p) cs += o[p][q];
            float r = 1.0f / cs;
#pragma unroll
            for (int p = 0; p < DEG; ++p) o[p][q] *= r;
        }
#pragma unroll
        for (int p = 0; p < DEG; ++p) {
            float rs = 1e-8f;
#pragma unroll
            for (int q = 0; q < DEG; ++q) rs += o[p][q];
            float r = 1.0f / rs;
#pragma unroll
            for (int q = 0; q < DEG; ++q) o[p][q] *= r;
        }
    }

    // ---- argmax per row (first max), gather S_hat at argmax, sum over p ----
    float cost = 0.0f;
#pragma unroll
    for (int p = 0; p < DEG; ++p) {
        float best = o[p][0];
        int bj = it[0];
#pragma unroll
        for (int q = 1; q < DEG; ++q) {
            if (o[p][q] > best) { best = o[p][q]; bj = it[q]; }
        }
        float sv = srows[p][bj];
        cost += (p < ds) ? sv : 0.0f;
    }

    int   M    = (ds > dt) ? ds : dt;
    float simv = sim[(size_t)(b * NS + i) * NT + j];
    float v    = simv + cost / (1.0f + (float)M);

    // ---- block softmax over j ----
    red[j] = v;
    __syncthreads();
    for (int off = NT / 2; off > 0; off >>= 1) {
        if (j < off) red[j] = fmaxf(red[j], red[j + off]);
        __syncthreads();
    }
    float m = red[0];
    __syncthreads();
    float e = __expf(v - m);
    red[j] = e;
    __syncthreads();
    for (int off = NT / 2; off > 0; off >>= 1) {
        if (j < off) red[j] += red[j + off];
        __syncthreads();
    }
    float ssum = red[0];
    out[(size_t)(b * NS + i) * NT + j] = e / ssum;
}

// ---------------------------------------------------------------------------
extern "C" void kernel_launch(void* const* d_in, const int* in_sizes, int n_in,
                              void* d_out, int out_size, void* d_ws, size_t ws_size,
                              hipStream_t stream)
{
    const float* xs = (const float*)d_in[0];
    const int*   es = (const int*)d_in[1];   // (2, B*NS*DEG)
    const float* xt = (const float*)d_in[3];
    const int*   et = (const int*)d_in[4];   // (2, B*NT*DEG)
    float*       outp = (float*)d_out;

    const int nEdgeS = BB * NS * DEG;
    const int nEdgeT = BB * NT * DEG;
    const int* dstS = es + nEdgeS;           // row 1 = dst
    const int* dstT = et + nEdgeT;

    float* sim  = (float*)d_ws;                                  // B*NS*NT f32
    int*   nbrS = (int*)((char*)d_ws + (size_t)BB * NS * NT * 4);
    int*   cntS = nbrS + BB * NS * DEG;
    int*   nbrT = cntS + BB * NS;
    int*   cntT = nbrT + BB * NT * DEG;

    nbr_kernel<<<(BB * NS + 127) / 128, 128, 0, stream>>>(dstS, nbrS, cntS, BB * NS);
    nbr_kernel<<<(BB * NT + 127) / 128, 128, 0, stream>>>(dstT, nbrT, cntT, BB * NT);
    sim_kernel<<<BB * 8, 256, 0, stream>>>(xs, xt, sim);
    pair_kernel<<<BB * NS, NT, 0, stream>>>(sim, nbrS, cntS, nbrT, cntT, outp);
}